// CasualAttention_48790828482902
// MI455X (gfx1250) — compile-verified
//
#include <hip/hip_runtime.h>
#include <math.h>
#include <stdint.h>

// ---------------------------------------------------------------------------
// Block-causal attention for MI455X (gfx1250), fp32 end-to-end using
// V_WMMA_F32_16X16X4_F32 (wave32 matrix pipe, exact-precision vs fp32 ref).
// Attention stages K/V through LDS with double-buffered CDNA5 async copies
// (global_load_async_to_lds_b128, pipelined via counted s_wait_asynccnt).
//
// Shapes: VP=8 B=2 C=512 H=W=16 -> T=2048, NH=8, hd=64, causal block=256.
// ---------------------------------------------------------------------------

typedef __attribute__((ext_vector_type(2))) float v2f;
typedef __attribute__((ext_vector_type(8))) float v8f;

#define VP     8
#define BATCH  2
#define CDIM   512
#define HDIM   16
#define WDIM   16
#define NHEAD  8
#define HD     64
#define TSEQ   2048   // VP*HDIM*WDIM
#define C3     1536   // 3*CDIM
#define CHW    (CDIM*HDIM*WDIM)   // 131072
#define KPAD   68     // LDS row stride (floats): 16B-aligned rows, no bank conflicts
#define KCHUNK 32     // keys staged per async chunk (per buffer)

// D = A(16x4,f32) x B(4x16,f32) + C(16x16,f32)
// A frag: lane m=lane&15 -> row M=m ; K = k0 + 2*(lane>=16) + {vgpr 0,1}
// B frag: lane m=lane&15 -> col N=m ; same K mapping
// C/D:    VGPR r -> rows {r, r+8} by lane half, N = lane&15
__device__ __forceinline__ v8f wmma_f32(v2f a, v2f b, v8f c) {
  return __builtin_amdgcn_wmma_f32_16x16x4_f32(
      /*neg_a=*/false, a, /*neg_b=*/false, b,
      /*c_mod=*/(short)0, c, /*reuse_a=*/false, /*reuse_b=*/false);
}

// CDNA5 async copy: 16B per lane, global -> LDS, tracked by ASYNCcnt.
__device__ __forceinline__ void async_load_b128(uint32_t lds_off,
                                                const float* gaddr) {
  asm volatile("global_load_async_to_lds_b128 %0, %1, off"
               :: "v"(lds_off), "v"((uint64_t)(uintptr_t)gaddr)
               : "memory");
}
template <int N>
__device__ __forceinline__ void wait_asynccnt() {
  asm volatile("s_wait_asynccnt %0" :: "n"(N) : "memory");
}

// ---------------------------------------------------------------------------
// Kernel 1: qkv = gather(x) @ Wqkv + bqkv -> q/k/v in [b][head][t][d] layout.
// One wave per 32(t) x 64(col) tile: per K-step 12 loads feed 8 WMMAs.
// t = v*256 + h*16 + w ; x flat offset = (v*B+b)*CHW + c*256 + h*16 + w.
// ---------------------------------------------------------------------------
__global__ __launch_bounds__(32) void qkv_proj_kernel(
    const float* __restrict__ X, const float* __restrict__ W,
    const float* __restrict__ bias, float* __restrict__ qs,
    float* __restrict__ ks, float* __restrict__ vs) {
  const int lane = threadIdx.x;
  const int half = lane >> 4;
  const int m    = lane & 15;
  const int n0   = blockIdx.x * 64;     // output column supertile in [0,1536)
  const int t0   = blockIdx.y * 32;     // token supertile
  const int b    = blockIdx.z;
  const int v    = t0 >> 8;
  const int h0   = (t0 >> 4) & 15;      // even
  const size_t xbase = (size_t)(v * BATCH + b) * CHW + (size_t)h0 * WDIM;

  v8f acc[2][4] = {};
#pragma unroll 4
  for (int k0 = 0; k0 < CDIM; k0 += 4) {
    const int kA = k0 + 2 * half;
    v2f a[2], bb[4];
#pragma unroll
    for (int mi = 0; mi < 2; ++mi) {
      a[mi].x = X[xbase + mi * WDIM + (size_t)kA * (HDIM * WDIM) + m];
      a[mi].y = X[xbase + mi * WDIM + (size_t)(kA + 1) * (HDIM * WDIM) + m];
    }
#pragma unroll
    for (int ni = 0; ni < 4; ++ni) {
      bb[ni].x = W[(size_t)kA * C3 + n0 + ni * 16 + m];
      bb[ni].y = W[(size_t)(kA + 1) * C3 + n0 + ni * 16 + m];
    }
#pragma unroll
    for (int mi = 0; mi < 2; ++mi)
#pragma unroll
      for (int ni = 0; ni < 4; ++ni)
        acc[mi][ni] = wmma_f32(a[mi], bb[ni], acc[mi][ni]);
  }

  // Scatter (n0 is 64-aligned -> supertile lies in one q/k/v slab and,
  // per 16-wide subtile, in one head).
#pragma unroll
  for (int ni = 0; ni < 4; ++ni) {
    const int col   = n0 + ni * 16 + m;
    const int which = col >> 9;        // 0:q 1:k 2:v
    const int cin   = col & 511;
    const int head  = cin >> 6;
    const int d     = cin & 63;
    float* dst = (which == 0) ? qs : (which == 1) ? ks : vs;
    const float bc = bias[col];
    const size_t hb = (size_t)(b * NHEAD + head) * TSEQ;
#pragma unroll
    for (int mi = 0; mi < 2; ++mi)
#pragma unroll
      for (int r = 0; r < 8; ++r) {
        const int t = t0 + mi * 16 + r + 8 * half;
        dst[(hb + t) * HD + d] = acc[mi][ni][r] + bc;
      }
  }
}

// ---------------------------------------------------------------------------
// Kernel 2: flash-style block-causal attention.
// 4 waves per block; wave w owns the 16-query tile qbase + 16*w (all 64
// queries share one causal block -> one shared key range). K/V are staged
// 32 keys at a time into double-buffered LDS with async b128 loads; each
// thread issues exactly 8 async ops per chunk, so s_wait_asynccnt 8 retires
// the older chunk while the newer chunk's DMA overlaps compute (async loads
// complete in order). Rows padded to 68 floats -> conflict-free strided
// fragment reads. Online softmax via shfl_xor {1,2,4,8} (stays inside
// 16-lane halves, matching the C/D row split M = r + 8*half). P transposed
// through per-wave LDS (DS ops are in-order within a wave -> no barrier).
// ---------------------------------------------------------------------------
__global__ __launch_bounds__(128) void attn_kernel(
    const float* __restrict__ qs, const float* __restrict__ ks,
    const float* __restrict__ vs, float* __restrict__ ys) {
  __shared__ float Klds[2][KCHUNK][KPAD];
  __shared__ float Vlds[2][KCHUNK][KPAD];
  __shared__ float Plds[4][16][17];

  const int tid  = threadIdx.x;
  const int wv   = tid >> 5;
  const int lane = tid & 31;
  const int half = lane >> 4;
  const int m    = lane & 15;
  const int head = blockIdx.y;
  const int b    = blockIdx.z;
  const int qbase = blockIdx.x * 64;
  const int tq0   = qbase + wv * 16;
  const size_t hb = (size_t)(b * NHEAD + head) * TSEQ;
  const float* Q = qs + hb * HD;
  const float* K = ks + hb * HD;
  const float* V = vs + hb * HD;

  // Stage chunk `ch` (32 keys of K and V) into LDS buffer `buf`.
  // 512 16B-chunks per panel; 128 threads x 4 chunks each -> 8 async/thread.
  auto stage = [&](int ch, int buf) {
#pragma unroll
    for (int i = 0; i < 4; ++i) {
      const int cid = tid + i * 128;
      const int row = cid >> 4;
      const int cp  = cid & 15;
      const uint32_t loff = (uint32_t)(row * KPAD * 4 + cp * 16);
      const size_t g = (size_t)(ch * KCHUNK + row) * HD + cp * 4;
      async_load_b128((uint32_t)(uintptr_t)(&Klds[buf][0][0]) + loff, K + g);
      async_load_b128((uint32_t)(uintptr_t)(&Vlds[buf][0][0]) + loff, V + g);
    }
  };

  // Q tile 16x64 as 16 A-fragments (resident all kernel).
  v2f qf[16];
#pragma unroll
  for (int j = 0; j < 16; ++j) {
    const int d = 4 * j + 2 * half;
    qf[j].x = Q[(size_t)(tq0 + m) * HD + d];
    qf[j].y = Q[(size_t)(tq0 + m) * HD + d + 1];
  }

  float mrun[8], lrun[8];
#pragma unroll
  for (int r = 0; r < 8; ++r) { mrun[r] = -INFINITY; lrun[r] = 0.0f; }
  v8f yacc[4] = {};

  const int nch = (((qbase >> 8) + 1) << 3);   // allowed 32-key chunks (>= 8)
  stage(0, 0);                                  // prologue
  for (int ch = 0; ch < nch; ++ch) {
    const int buf = ch & 1;
    if (ch + 1 < nch) {
      stage(ch + 1, buf ^ 1);   // target buffer was released by the previous barrier
      wait_asynccnt<8>();       // retire chunk `ch` (in-order completion)
    } else {
      wait_asynccnt<0>();
    }
    __syncthreads();

    // ---- 2 key subtiles of 16 ----
#pragma unroll
    for (int st = 0; st < 2; ++st) {
      const int kb = st * 16;

      // S = Q x K^T (B-frag: lane m -> key kb+m, halves/vgprs -> d).
      v8f s = {};
#pragma unroll
      for (int j = 0; j < 16; ++j) {
        const int d = 4 * j + 2 * half;
        v2f kf;
        kf.x = Klds[buf][kb + m][d];
        kf.y = Klds[buf][kb + m][d + 1];
        s = wmma_f32(qf[j], kf, s);
      }

      // Online softmax update; row M = r + 8*half in this lane half.
#pragma unroll
      for (int r = 0; r < 8; ++r) {
        float sv = s[r] * 0.125f;   // 1/sqrt(64)
        float rm = sv;
        rm = fmaxf(rm, __shfl_xor(rm, 1));
        rm = fmaxf(rm, __shfl_xor(rm, 2));
        rm = fmaxf(rm, __shfl_xor(rm, 4));
        rm = fmaxf(rm, __shfl_xor(rm, 8));
        const float mnew = fmaxf(mrun[r], rm);
        const float p    = __expf(sv - mnew);
        float rs = p;
        rs += __shfl_xor(rs, 1);
        rs += __shfl_xor(rs, 2);
        rs += __shfl_xor(rs, 4);
        rs += __shfl_xor(rs, 8);
        const float alpha = __expf(mrun[r] - mnew);
        lrun[r] = lrun[r] * alpha + rs;
        mrun[r] = mnew;
#pragma unroll
        for (int nt = 0; nt < 4; ++nt) yacc[nt][r] *= alpha;
        Plds[wv][r + 8 * half][m] = p;
      }

      // P back in A-fragment layout (same wave, DS in-order).
      v2f pf[4];
#pragma unroll
      for (int j = 0; j < 4; ++j) {
        const int kk = 4 * j + 2 * half;
        pf[j].x = Plds[wv][m][kk];
        pf[j].y = Plds[wv][m][kk + 1];
      }

      // Y += P (16x16) x V (16x64).
#pragma unroll
      for (int nt = 0; nt < 4; ++nt) {
#pragma unroll
        for (int j = 0; j < 4; ++j) {
          const int kk = 4 * j + 2 * half;
          v2f vf;
          vf.x = Vlds[buf][kb + kk][nt * 16 + m];
          vf.y = Vlds[buf][kb + kk + 1][nt * 16 + m];
          yacc[nt] = wmma_f32(pf[j], vf, yacc[nt]);
        }
      }
    }
    __syncthreads();   // release `buf` for the chunk after next
  }

  // Normalize and store to ys[b][t][c], c = head*64 + nt*16 + m.
#pragma unroll
  for (int r = 0; r < 8; ++r) {
    const float inv = 1.0f / lrun[r];
    const int t = tq0 + r + 8 * half;
#pragma unroll
    for (int nt = 0; nt < 4; ++nt)
      ys[((size_t)b * TSEQ + t) * CDIM + head * HD + nt * 16 + m] =
          yacc[nt][r] * inv;
  }
}

// ---------------------------------------------------------------------------
// Kernel 3: out = scatter( ys @ Wo + bo ) back to [v][b][c][h][w].
// One wave per 32(t) x 64(c) tile (same scheme as kernel 1).
// ---------------------------------------------------------------------------
__global__ __launch_bounds__(32) void out_proj_kernel(
    const float* __restrict__ ys, const float* __restrict__ Wo,
    const float* __restrict__ bo, float* __restrict__ out) {
  const int lane = threadIdx.x;
  const int half = lane >> 4;
  const int m    = lane & 15;
  const int n0   = blockIdx.x * 64;   // output channel supertile
  const int t0   = blockIdx.y * 32;   // token supertile
  const int b    = blockIdx.z;
  const int v    = t0 >> 8;
  const int h0   = (t0 >> 4) & 15;    // even

  v8f acc[2][4] = {};
#pragma unroll 4
  for (int k0 = 0; k0 < CDIM; k0 += 4) {
    const int kA = k0 + 2 * half;
    v2f a[2], bb[4];
#pragma unroll
    for (int mi = 0; mi < 2; ++mi) {
      const size_t rowb = ((size_t)b * TSEQ + t0 + mi * 16 + m) * CDIM;
      a[mi].x = ys[rowb + kA];
      a[mi].y = ys[rowb + kA + 1];
    }
#pragma unroll
    for (int ni = 0; ni < 4; ++ni) {
      bb[ni].x = Wo[(size_t)kA * CDIM + n0 + ni * 16 + m];
      bb[ni].y = Wo[(size_t)(kA + 1) * CDIM + n0 + ni * 16 + m];
    }
#pragma unroll
    for (int mi = 0; mi < 2; ++mi)
#pragma unroll
      for (int ni = 0; ni < 4; ++ni)
        acc[mi][ni] = wmma_f32(a[mi], bb[ni], acc[mi][ni]);
  }

#pragma unroll
  for (int mi = 0; mi < 2; ++mi) {
    const size_t obase =
        (size_t)(v * BATCH + b) * CHW + (size_t)(h0 + mi) * WDIM;
#pragma unroll
    for (int ni = 0; ni < 4; ++ni) {
      const int c = n0 + ni * 16 + m;
      const float bc = bo[c];
#pragma unroll
      for (int r = 0; r < 8; ++r) {
        const int w = r + 8 * half;   // row M = w
        out[obase + (size_t)c * (HDIM * WDIM) + w] = acc[mi][ni][r] + bc;
      }
    }
  }
}

// ---------------------------------------------------------------------------
extern "C" void kernel_launch(void* const* d_in, const int* in_sizes, int n_in,
                              void* d_out, int out_size, void* d_ws, size_t ws_size,
                              hipStream_t stream) {
  (void)in_sizes; (void)n_in; (void)out_size; (void)ws_size;
  const float* x    = (const float*)d_in[0];
  const float* Wqkv = (const float*)d_in[1];
  const float* bqkv = (const float*)d_in[2];
  const float* Wo   = (const float*)d_in[3];
  const float* bo   = (const float*)d_in[4];
  float* out = (float*)d_out;

  float* ws = (float*)d_ws;
  const size_t SZ = (size_t)BATCH * NHEAD * TSEQ * HD;   // 2,097,152 floats
  float* qs = ws;
  float* ks = ws + SZ;
  float* vs = ws + 2 * SZ;
  float* ys = ws + 3 * SZ;   // [B][T][C], same element count

  qkv_proj_kernel<<<dim3(C3 / 64, TSEQ / 32, BATCH), dim3(32, 1, 1), 0,
                    stream>>>(x, Wqkv, bqkv, qs, ks, vs);
  attn_kernel<<<dim3(TSEQ / 64, NHEAD, BATCH), dim3(128, 1, 1), 0, stream>>>(
      qs, ks, vs, ys);
  out_proj_kernel<<<dim3(CDIM / 64, TSEQ / 32, BATCH), dim3(32, 1, 1), 0,
                    stream>>>(ys, Wo, bo, out);
}